// MonoModel_584115552925
// MI455X (gfx1250) — compile-verified
//
#include <hip/hip_runtime.h>
#include <cstdint>
#include <cstddef>

typedef __attribute__((ext_vector_type(2))) float v2f;
typedef __attribute__((ext_vector_type(8))) float v8f;

#define F_IN 256
#define C1   64
#define C2   40
#define C2P  48   // padded to 3 WMMA n-tiles

// ---------------- degree / dinv ----------------
__global__ void k_zero(float* __restrict__ p, int n) {
  int i = blockIdx.x * blockDim.x + threadIdx.x;
  if (i < n) p[i] = 0.0f;
}

__global__ void k_deg(const long long* __restrict__ dst, long long E, float* __restrict__ deg) {
  long long i = (long long)blockIdx.x * blockDim.x + threadIdx.x;
  if (i < E) atomicAdd(&deg[(int)dst[i]], 1.0f);
}

__global__ void k_dinv(float* __restrict__ p, int n) {
  int i = blockIdx.x * blockDim.x + threadIdx.x;
  if (i < n) p[i] = rsqrtf(p[i] + 1.0f);  // +1 = self loop; deg >= 1 always
}

// ---------------- GEMM1: [N,256] @ [256,64] via V_WMMA_F32_16X16X4_F32 ----------------
// Block: 128 threads = 4 wave32; wave w computes C tile rows [16*bx,16*bx+16), cols [16w,16w+16).
// fp32 WMMA fragment layouts (ISA 7.12.2):
//   A 16x4 : lane l holds A[l%16][2*(l/16)+v], v=0,1
//   B 4x16 : lane l holds B[2*(l/16)+v][l%16]
//   C 16x16: lane l holds C[v+8*(l/16)][l%16], v=0..7
__global__ void __launch_bounds__(128)
k_gemm1(const float* __restrict__ X, const float* __restrict__ W, float* __restrict__ Y) {
  __shared__ float WT[C1 * 66];  // W-slab transposed [n][k], row-padded to 66 (bank-conflict free)
  const int tid  = threadIdx.x;
  const int lane = tid & 31;
  const int wave = tid >> 5;       // n-tile
  const int mr   = lane & 15;
  const int half = lane >> 4;
  const int m0   = blockIdx.x * 16;
  const int n    = wave * 16 + mr;

  v8f acc = {};
  const float* xrow = X + (size_t)(m0 + mr) * F_IN + 2 * half;

  for (int ks = 0; ks < F_IN; ks += 64) {
    for (int i = tid; i < 64 * C1; i += 128) {
      int nn = i & (C1 - 1);
      int kk = i >> 6;
      WT[nn * 66 + kk] = W[(size_t)(ks + kk) * C1 + nn];
    }
    __syncthreads();
#pragma unroll
    for (int kk = 0; kk < 64; kk += 4) {
      v2f a = *(const v2f*)(xrow + ks + kk);
      v2f b = *(const v2f*)(&WT[n * 66 + kk + 2 * half]);
      acc = __builtin_amdgcn_wmma_f32_16x16x4_f32(false, a, false, b, (short)0, acc, false, false);
    }
    __syncthreads();
  }
#pragma unroll
  for (int v = 0; v < 8; ++v)
    Y[(size_t)(m0 + v + 8 * half) * C1 + n] = acc[v];
}

// ---------------- GEMM2: [N,64] @ [64,40] (padded to 48) ----------------
__global__ void __launch_bounds__(96)
k_gemm2(const float* __restrict__ X, const float* __restrict__ W, float* __restrict__ Y) {
  __shared__ float WT[C2P * 66];
  const int tid  = threadIdx.x;
  const int lane = tid & 31;
  const int wave = tid >> 5;       // 0..2
  const int mr   = lane & 15;
  const int half = lane >> 4;
  const int m0   = blockIdx.x * 16;
  const int n    = wave * 16 + mr; // 0..47

  for (int i = tid; i < C2P * C1; i += 96) {
    int nn = i % C2P;
    int kk = i / C2P;
    WT[nn * 66 + kk] = (nn < C2) ? W[(size_t)kk * C2 + nn] : 0.0f;
  }
  __syncthreads();

  v8f acc = {};
  const float* xrow = X + (size_t)(m0 + mr) * C1 + 2 * half;
#pragma unroll
  for (int kk = 0; kk < C1; kk += 4) {
    v2f a = *(const v2f*)(xrow + kk);
    v2f b = *(const v2f*)(&WT[n * 66 + kk + 2 * half]);
    acc = __builtin_amdgcn_wmma_f32_16x16x4_f32(false, a, false, b, (short)0, acc, false, false);
  }
  if (n < C2) {
#pragma unroll
    for (int v = 0; v < 8; ++v)
      Y[(size_t)(m0 + v + 8 * half) * C2 + n] = acc[v];
  }
}

// ---------------- aggregation ----------------
template <int C>
__global__ void k_selfloop(const float* __restrict__ xw, const float* __restrict__ dinv,
                           float* __restrict__ agg, long long total) {
  long long i = (long long)blockIdx.x * blockDim.x + threadIdx.x;
  if (i < total) {
    int node = (int)(i / C);
    float w = dinv[node];
    agg[i] = xw[i] * w * w;   // self-loop term; also initializes the buffer
  }
}

template <int C, int G>  // G = C/4 float4 groups per edge
__global__ void k_scatter(const long long* __restrict__ src, const long long* __restrict__ dst,
                          const float* __restrict__ dinv, const float* __restrict__ xw,
                          float* __restrict__ agg, long long E) {
  long long t = (long long)blockIdx.x * blockDim.x + threadIdx.x;
  long long e = t / G;
  int g = (int)(t - e * G);
  if (e >= E) return;
  int s = (int)src[e];
  int d = (int)dst[e];
  float w = dinv[s] * dinv[d];
  const float4 v = *(const float4*)(xw + (size_t)s * C + g * 4);
  float* o = agg + (size_t)d * C + g * 4;
  atomicAdd(o + 0, v.x * w);
  atomicAdd(o + 1, v.y * w);
  atomicAdd(o + 2, v.z * w);
  atomicAdd(o + 3, v.w * w);
}

__global__ void k_bias_relu(float* __restrict__ h, const float* __restrict__ b, long long total) {
  long long i = (long long)blockIdx.x * blockDim.x + threadIdx.x;
  if (i < total) h[i] = fmaxf(h[i] + b[i & (C1 - 1)], 0.0f);
}

__global__ void k_logsoftmax(const float* __restrict__ agg, const float* __restrict__ b,
                             float* __restrict__ out, int N) {
  int node = blockIdx.x * blockDim.x + threadIdx.x;
  if (node >= N) return;
  const float* row = agg + (size_t)node * C2;
  float vals[C2];
  float m = -3.402823466e38f;
#pragma unroll
  for (int c = 0; c < C2; ++c) { vals[c] = row[c] + b[c]; m = fmaxf(m, vals[c]); }
  float s = 0.0f;
#pragma unroll
  for (int c = 0; c < C2; ++c) s += expf(vals[c] - m);
  float ls = m + logf(s);
  float* orow = out + (size_t)node * C2;
#pragma unroll
  for (int c = 0; c < C2; ++c) orow[c] = vals[c] - ls;
}

// ---------------- launcher ----------------
extern "C" void kernel_launch(void* const* d_in, const int* in_sizes, int n_in,
                              void* d_out, int out_size, void* d_ws, size_t ws_size,
                              hipStream_t stream) {
  (void)n_in; (void)out_size; (void)ws_size;
  const float*     x   = (const float*)d_in[0];
  const long long* ei  = (const long long*)d_in[1];   // int64 [2,E]
  const float*     W1  = (const float*)d_in[2];
  const float*     b1  = (const float*)d_in[3];
  const float*     W2  = (const float*)d_in[4];
  const float*     b2  = (const float*)d_in[5];
  const int        N   = in_sizes[0] / F_IN;          // 100000 (divisible by 16)
  const long long  E   = (long long)in_sizes[1] / 2;
  const long long* src = ei;
  const long long* dst = ei + E;

  char* wsp = (char*)d_ws;
  auto alloc = [&](size_t bytes) -> float* {
    float* p = (float*)wsp;
    wsp += (bytes + 255) & ~(size_t)255;
    return p;
  };
  float* dinv = alloc((size_t)N * 4);            // deg, then dinv in-place
  float* xw1  = alloc((size_t)N * C1 * 4);
  float* h1   = alloc((size_t)N * C1 * 4);       // agg1, then relu'd in-place
  float* xw2  = alloc((size_t)N * C2 * 4);
  float* agg2 = alloc((size_t)N * C2 * 4);
  float* out  = (float*)d_out;

  const int B = 256;
  // degrees (recomputed every call: workspace is not preserved/zeroed between replays)
  k_zero<<<(N + B - 1) / B, B, 0, stream>>>(dinv, N);
  k_deg<<<(unsigned)((E + B - 1) / B), B, 0, stream>>>(dst, E, dinv);
  k_dinv<<<(N + B - 1) / B, B, 0, stream>>>(dinv, N);

  // layer 1
  k_gemm1<<<N / 16, 128, 0, stream>>>(x, W1, xw1);
  const long long n1 = (long long)N * C1;
  k_selfloop<C1><<<(unsigned)((n1 + B - 1) / B), B, 0, stream>>>(xw1, dinv, h1, n1);
  const long long t1 = E * (C1 / 4);
  k_scatter<C1, C1 / 4><<<(unsigned)((t1 + B - 1) / B), B, 0, stream>>>(src, dst, dinv, xw1, h1, E);
  k_bias_relu<<<(unsigned)((n1 + B - 1) / B), B, 0, stream>>>(h1, b1, n1);

  // layer 2
  k_gemm2<<<N / 16, 96, 0, stream>>>(h1, W2, xw2);
  const long long n2 = (long long)N * C2;
  k_selfloop<C2><<<(unsigned)((n2 + B - 1) / B), B, 0, stream>>>(xw2, dinv, agg2, n2);
  const long long t2 = E * (C2 / 4);
  k_scatter<C2, C2 / 4><<<(unsigned)((t2 + B - 1) / B), B, 0, stream>>>(src, dst, dinv, xw2, agg2, E);

  k_logsoftmax<<<(N + B - 1) / B, B, 0, stream>>>(agg2, b2, out, N);
}